// GAT_53618371723353
// MI455X (gfx1250) — compile-verified
//
#include <hip/hip_runtime.h>

#define NNODES   50000
#define NEDGES   800000
#define IN_DIM   128
#define HID_DIM  32
#define OUT_DIM  64
#define NHEADS   2
#define NEG_SLOPE 0.2f

typedef __attribute__((ext_vector_type(2))) float v2f;
typedef __attribute__((ext_vector_type(8))) float v8f;

// ---------------------------------------------------------------------------
// GEMM: Y[N,F] = X[N,K] @ W[K,F], fp32 WMMA 16x16x4. One wave per 16x16 tile.
// A layout (16x4 f32, ISA 7.12.2): lane lo=lane&15 holds row M=lo; hi=lane>>4
//   selects K pair: VGPR0=K(2*hi), VGPR1=K(2*hi+1).
// B layout (4x16): lane lo holds col N=lo; VGPR0=K(2*hi), VGPR1=K(2*hi+1).
// D layout: VGPR i -> row (i + 8*hi), col lo.
// ---------------------------------------------------------------------------
__global__ __launch_bounds__(256) void gemm_wmma_f32(
    const float* __restrict__ X, const float* __restrict__ W,
    float* __restrict__ Y, int Nrows, int K, int F)
{
  const int lane  = threadIdx.x & 31;
  const int wave  = blockIdx.x * (blockDim.x >> 5) + (threadIdx.x >> 5);
  const int tilesF = F >> 4;
  const int tileM  = wave / tilesF;
  const int tileF  = wave - tileM * tilesF;
  if (tileM * 16 >= Nrows) return;           // whole wave exits together
  const int row0 = tileM << 4;
  const int col0 = tileF << 4;
  const int lo = lane & 15;
  const int hi = lane >> 4;

  v8f acc = {};
  const float* ap = X + (size_t)(row0 + lo) * K + 2 * hi;
  const float* bp = W + (size_t)(2 * hi) * F + col0 + lo;
  for (int k = 0; k < K; k += 4) {
    v2f a; a.x = ap[0]; a.y = ap[1];
    v2f b; b.x = bp[0]; b.y = bp[F];
    ap += 4;
    bp += (size_t)4 * F;
    acc = __builtin_amdgcn_wmma_f32_16x16x4_f32(
        /*neg_a=*/false, a, /*neg_b=*/false, b,
        /*c_mod=*/(short)0, acc, /*reuse_a=*/false, /*reuse_b=*/false);
  }
#pragma unroll
  for (int i = 0; i < 8; i++)
    Y[(size_t)(row0 + i + 8 * hi) * F + col0 + lo] = acc[i];
}

// ---------------------------------------------------------------------------
// el/er attention scores: one thread per (node, head)
// ---------------------------------------------------------------------------
__global__ __launch_bounds__(256) void attn_scores(
    const float* __restrict__ feat, const float* __restrict__ al,
    const float* __restrict__ ar, float* __restrict__ el,
    float* __restrict__ er, int D, int total)
{
  int i = blockIdx.x * blockDim.x + threadIdx.x;
  if (i >= total) return;
  int n = i >> 1, h = i & 1;                 // NHEADS == 2
  const float* f = feat + (size_t)n * (NHEADS * D) + h * D;
  const float* a = al + h * D;
  const float* r = ar + h * D;
  float sl = 0.0f, sr = 0.0f;
  for (int d = 0; d < D; d++) { float v = f[d]; sl += v * a[d]; sr += v * r[d]; }
  el[i] = sl; er[i] = sr;
}

// ---------------------------------------------------------------------------
// Per-layer scratch init: m=-inf, s=0, agg=0
// ---------------------------------------------------------------------------
__global__ __launch_bounds__(256) void init_layer(
    float* __restrict__ m, float* __restrict__ s, float* __restrict__ agg,
    int nh, int nf)
{
  int i = blockIdx.x * blockDim.x + threadIdx.x;
  if (i < nf) agg[i] = 0.0f;
  if (i < nh) { m[i] = -__builtin_inff(); s[i] = 0.0f; }
}

// float atomic max via sign-aware integer atomics (handles -0, mixed signs)
__device__ __forceinline__ void atomicMaxF(float* addr, float v) {
  unsigned u = __float_as_uint(v);
  if (u >> 31) atomicMin((unsigned int*)addr, u);
  else         atomicMax((int*)addr, (int)u);
}

// ---------------------------------------------------------------------------
// Edge pass A: leaky-relu logit, store, segment-max into m[dst]
// ---------------------------------------------------------------------------
__global__ __launch_bounds__(256) void edge_logits_max(
    const int* __restrict__ src, const int* __restrict__ dst,
    const float* __restrict__ el, const float* __restrict__ er,
    float* __restrict__ ee, float* __restrict__ m, int total)
{
  int i = blockIdx.x * blockDim.x + threadIdx.x;
  if (i >= total) return;
  int e = i >> 1, h = i & 1;
  int si = src[e], di = dst[e];
  float v = el[si * NHEADS + h] + er[di * NHEADS + h];
  v = (v > 0.0f) ? v : NEG_SLOPE * v;
  ee[i] = v;
  atomicMaxF(&m[di * NHEADS + h], v);
}

// ---------------------------------------------------------------------------
// Edge pass B: ex = exp(e - m[dst]) stored in place; segment-sum into s[dst]
// ---------------------------------------------------------------------------
__global__ __launch_bounds__(256) void edge_exp_sum(
    const int* __restrict__ dst, float* __restrict__ ee,
    const float* __restrict__ m, float* __restrict__ s, int total)
{
  int i = blockIdx.x * blockDim.x + threadIdx.x;
  if (i >= total) return;
  int e = i >> 1, h = i & 1;
  int di = dst[e];
  float ex = expf(ee[i] - m[di * NHEADS + h]);
  ee[i] = ex;
  atomicAdd(&s[di * NHEADS + h], ex);
}

// ---------------------------------------------------------------------------
// Edge pass C: agg[dst, f] += feat[src, f] * ex  (one thread per edge-feature)
// ---------------------------------------------------------------------------
__global__ __launch_bounds__(256) void edge_aggregate(
    const int* __restrict__ src, const int* __restrict__ dst,
    const float* __restrict__ ee, const float* __restrict__ feat,
    float* __restrict__ agg, int fshift, int dshift, long total)
{
  long i = (long)blockIdx.x * blockDim.x + threadIdx.x;
  if (i >= total) return;
  int e = (int)(i >> fshift);
  int f = (int)(i - ((long)e << fshift));
  int h = f >> dshift;
  float ex = ee[e * NHEADS + h];
  atomicAdd(&agg[((size_t)dst[e] << fshift) + f],
            feat[((size_t)src[e] << fshift) + f] * ex);
}

// ---------------------------------------------------------------------------
// Node finish (layers 0/1): out = relu(agg/s + b); s==0 -> no in-edges -> 0+b
// ---------------------------------------------------------------------------
__global__ __launch_bounds__(256) void node_finish(
    const float* __restrict__ agg, const float* __restrict__ s,
    const float* __restrict__ bias, float* __restrict__ out,
    int fshift, int dshift, int total, int relu)
{
  int i = blockIdx.x * blockDim.x + threadIdx.x;
  if (i >= total) return;
  int n = i >> fshift;
  int f = i - (n << fshift);
  int h = f >> dshift;
  float ss = s[n * NHEADS + h];
  float v = (ss > 0.0f) ? agg[i] / ss : 0.0f;
  v += bias[f];
  if (relu) v = fmaxf(v, 0.0f);
  out[i] = v;
}

// ---------------------------------------------------------------------------
// Final layer: mean over heads -> logits [N, OUT_DIM]
// ---------------------------------------------------------------------------
__global__ __launch_bounds__(256) void node_final_mean(
    const float* __restrict__ agg, const float* __restrict__ s,
    const float* __restrict__ bias, float* __restrict__ out, int total)
{
  int i = blockIdx.x * blockDim.x + threadIdx.x;
  if (i >= total) return;
  int n = i / OUT_DIM;
  int o = i - n * OUT_DIM;
  float s0 = s[n * 2 + 0], s1 = s[n * 2 + 1];
  size_t base = (size_t)n * 2 * OUT_DIM;
  float v0 = ((s0 > 0.0f) ? agg[base + o] / s0 : 0.0f) + bias[o];
  float v1 = ((s1 > 0.0f) ? agg[base + OUT_DIM + o] / s1 : 0.0f) + bias[OUT_DIM + o];
  out[i] = 0.5f * (v0 + v1);
}

// ---------------------------------------------------------------------------
extern "C" void kernel_launch(void* const* d_in, const int* in_sizes, int n_in,
                              void* d_out, int out_size, void* d_ws, size_t ws_size,
                              hipStream_t stream) {
  const float* x   = (const float*)d_in[0];
  const int*   src = (const int*)  d_in[1];
  const int*   dst = (const int*)  d_in[2];
  const float* Wm[3]  = { (const float*)d_in[3],  (const float*)d_in[7],  (const float*)d_in[11] };
  const float* alv[3] = { (const float*)d_in[4],  (const float*)d_in[8],  (const float*)d_in[12] };
  const float* arv[3] = { (const float*)d_in[5],  (const float*)d_in[9],  (const float*)d_in[13] };
  const float* bv[3]  = { (const float*)d_in[6],  (const float*)d_in[10], (const float*)d_in[14] };

  // workspace layout (floats)
  float* feat_a = (float*)d_ws;                               // N*128
  float* feat_b = feat_a + (size_t)NNODES * 128;              // N*128
  float* agg    = feat_b + (size_t)NNODES * 128;              // N*128
  float* el     = agg    + (size_t)NNODES * 128;              // N*2
  float* er     = el     + (size_t)NNODES * NHEADS;           // N*2
  float* mbuf   = er     + (size_t)NNODES * NHEADS;           // N*2
  float* sbuf   = mbuf   + (size_t)NNODES * NHEADS;           // N*2
  float* ee     = sbuf   + (size_t)NNODES * NHEADS;           // E*2

  const int B = 256;
  auto blocks  = [](long n) { return (unsigned)((n + 255) / 256); };

  auto run_layer = [&](const float* X, int K, int layer, int D, float* featbuf,
                       float* outbuf, bool relu, bool final_mean) {
    const int F = NHEADS * D;                 // 64 or 128
    const int dshift = (D == 32) ? 5 : 6;
    const int fshift = dshift + 1;

    // feature transform (WMMA)
    const long waves = (long)(NNODES / 16) * (F / 16);
    gemm_wmma_f32<<<blocks(waves * 32), B, 0, stream>>>(X, Wm[layer], featbuf,
                                                        NNODES, K, F);
    // attention scores
    attn_scores<<<blocks((long)NNODES * NHEADS), B, 0, stream>>>(
        featbuf, alv[layer], arv[layer], el, er, D, NNODES * NHEADS);
    // init m/s/agg
    init_layer<<<blocks((long)NNODES * F), B, 0, stream>>>(
        mbuf, sbuf, agg, NNODES * NHEADS, NNODES * F);
    // edge softmax (max, exp+sum), then weighted scatter
    edge_logits_max<<<blocks((long)NEDGES * NHEADS), B, 0, stream>>>(
        src, dst, el, er, ee, mbuf, NEDGES * NHEADS);
    edge_exp_sum<<<blocks((long)NEDGES * NHEADS), B, 0, stream>>>(
        dst, ee, mbuf, sbuf, NEDGES * NHEADS);
    const long tot = (long)NEDGES * F;
    edge_aggregate<<<blocks(tot), B, 0, stream>>>(src, dst, ee, featbuf, agg,
                                                  fshift, dshift, tot);
    // node epilogue
    if (!final_mean) {
      node_finish<<<blocks((long)NNODES * F), B, 0, stream>>>(
          agg, sbuf, bv[layer], outbuf, fshift, dshift, NNODES * F, relu ? 1 : 0);
    } else {
      node_final_mean<<<blocks((long)NNODES * OUT_DIM), B, 0, stream>>>(
          agg, sbuf, bv[layer], outbuf, NNODES * OUT_DIM);
    }
  };

  // layer 0: x[N,128] -> feat_a -> aggregate -> feat_b [N,64]
  run_layer(x,      IN_DIM,          0, HID_DIM, feat_a, feat_b, true,  false);
  // layer 1: feat_b[N,64] -> feat_a -> aggregate -> feat_b [N,64]
  run_layer(feat_b, NHEADS * HID_DIM, 1, HID_DIM, feat_a, feat_b, true,  false);
  // layer 2: feat_b[N,64] -> feat_a[N,128] -> aggregate -> mean heads -> d_out
  run_layer(feat_b, NHEADS * HID_DIM, 2, OUT_DIM, feat_a, (float*)d_out, false, true);
}